// Loss_37735582663369
// MI455X (gfx1250) — compile-verified
//
#include <hip/hip_runtime.h>
#include <math.h>

// SSD MultiBox loss for MI455X (gfx1250, wave32). Bandwidth-bound (~100 MB
// traffic, ~90 MFLOP -> ~4.3us floor at 23.3 TB/s). All global streams are
// 128-bit vectorized (NB=8732 divisible by 4; channel strides 16B-aligned).
// Kernel 1: one 1024-thread block per row; elementwise losses + u32 keys in
//           LDS + 4-level radix select for hard-negative mining.
// Kernel 2: 256->1 mean via chained V_WMMA_F32_16X16X4_F32 (A = ones).

#define NBATCH 256
#define NBOX   8732
#define NBQ    (NBOX / 4)      // 2183 float4 packets per stream
#define TPB    1024
#define NWAVES (TPB / 32)

typedef float v2f __attribute__((ext_vector_type(2)));
typedef float v8f __attribute__((ext_vector_type(8)));

__device__ __forceinline__ float waveSumF(float v) {
    v += __shfl_xor(v, 16, 32);
    v += __shfl_xor(v, 8, 32);
    v += __shfl_xor(v, 4, 32);
    v += __shfl_xor(v, 2, 32);
    v += __shfl_xor(v, 1, 32);
    return v;
}
__device__ __forceinline__ int waveSumI(int v) {
    v += __shfl_xor(v, 16, 32);
    v += __shfl_xor(v, 8, 32);
    v += __shfl_xor(v, 4, 32);
    v += __shfl_xor(v, 2, 32);
    v += __shfl_xor(v, 1, 32);
    return v;
}

__device__ __forceinline__ float fcomp(const float4& v, int j) {
    return ((const float*)&v)[j];
}
__device__ __forceinline__ int icomp(const int4& v, int j) {
    return ((const int*)&v)[j];
}

__global__ __launch_bounds__(TPB) void ssd_row_loss(
    const float* __restrict__ ploc,    // [N,4,NB]
    const float* __restrict__ plabel,  // [N,1,NB]
    const float* __restrict__ gloc,    // [N,4,NB]
    const int*   __restrict__ glabel,  // [N,NB]
    const float* __restrict__ dboxes,  // [1,4,NB]
    float*       __restrict__ row_out) // [N]
{
    __shared__ uint4    keys4[NBQ];    // con_neg bit patterns, 4 per packet
    __shared__ unsigned hist[256];
    __shared__ float    fscrA[NWAVES];
    __shared__ float    fscrB[NWAVES];
    __shared__ int      iscr[NWAVES];
    __shared__ float    s_sl1, s_poscon;
    __shared__ int      s_pos;
    __shared__ unsigned s_prefix;
    __shared__ int      s_remaining;

    const int n    = blockIdx.x;
    const int tid  = threadIdx.x;
    const int lane = tid & 31;
    const int wave = tid >> 5;

    if (tid < 256) hist[tid] = 0u;
    __syncthreads();

    // 128-bit packetized views of every stream (all 16B-aligned: NBOX%4==0)
    const float4* pl0 = (const float4*)(ploc   + (size_t)n * 4 * NBOX);
    const float4* pl1 = pl0 + NBQ;
    const float4* pl2 = pl0 + 2 * NBQ;
    const float4* pl3 = pl0 + 3 * NBQ;
    const float4* gl0 = (const float4*)(gloc   + (size_t)n * 4 * NBOX);
    const float4* gl1 = gl0 + NBQ;
    const float4* gl2 = gl0 + 2 * NBQ;
    const float4* gl3 = gl0 + 3 * NBQ;
    const float4* pb4 = (const float4*)(plabel + (size_t)n * NBOX);
    const int4*   lb4 = (const int4*)(glabel + (size_t)n * NBOX);
    const float4* db0 = (const float4*)dboxes;
    const float4* db1 = db0 + NBQ;
    const float4* db2 = db0 + 2 * NBQ;
    const float4* db3 = db0 + 3 * NBQ;

    float sl1_acc = 0.f, poscon_acc = 0.f;
    int   pos_acc = 0;

    // ---- Pass A: elementwise losses + key build + level-0 histogram ----
    for (int q = tid; q < NBQ; q += TPB) {
        if (q + TPB < NBQ) {  // gfx1250 speculative global_prefetch_b8
            __builtin_prefetch(&pl0[q + TPB], 0, 0);
            __builtin_prefetch(&gl0[q + TPB], 0, 0);
        }
        const int4   L  = lb4[q];
        const float4 X  = pb4[q];
        const float4 P0 = pl0[q], P1 = pl1[q], P2 = pl2[q], P3 = pl3[q];
        const float4 G0 = gl0[q], G1 = gl1[q], G2 = gl2[q], G3 = gl3[q];
        const float4 D0 = db0[q], D1 = db1[q], D2 = db2[q], D3 = db3[q];

        uint4 kk;
        unsigned* kv = (unsigned*)&kk;
#pragma unroll
        for (int j = 0; j < 4; ++j) {
            const int   g = icomp(L, j);
            const bool  m = g > 0;
            const float y = (float)g;
            const float x = fcomp(X, j);
            // numerically stable BCE-with-logits (>= 0 when y in {0,1})
            const float con = fmaxf(x, 0.f) - x * y + log1pf(expf(-fabsf(x)));

            const float w = fcomp(D2, j), h = fcomp(D3, j);
            const float v0 = (fcomp(G0, j) - fcomp(D0, j)) / w;
            const float v1 = (fcomp(G1, j) - fcomp(D1, j)) / h;
            const float v2 = logf(fcomp(G2, j) / w);
            const float v3 = logf(fcomp(G3, j) / h);

            float s = 0.f, d, ad;
            d = fcomp(P0, j) - v0; ad = fabsf(d);
            s += (ad < 1.f) ? 0.5f * d * d : ad - 0.5f;
            d = fcomp(P1, j) - v1; ad = fabsf(d);
            s += (ad < 1.f) ? 0.5f * d * d : ad - 0.5f;
            d = fcomp(P2, j) - v2; ad = fabsf(d);
            s += (ad < 1.f) ? 0.5f * d * d : ad - 0.5f;
            d = fcomp(P3, j) - v3; ad = fabsf(d);
            s += (ad < 1.f) ? 0.5f * d * d : ad - 0.5f;

            if (m) { pos_acc++; sl1_acc += s; poscon_acc += con; }

            const unsigned key = m ? 0u : __float_as_uint(fmaxf(con, 0.f));
            kv[j] = key;
            atomicAdd(&hist[key >> 24], 1u);
        }
        keys4[q] = kk;   // ds_store_b128
    }

    // ---- block reduction (fixed order -> deterministic) ----
    {
        float a = waveSumF(sl1_acc);
        float b = waveSumF(poscon_acc);
        int   p = waveSumI(pos_acc);
        if (lane == 0) { fscrA[wave] = a; fscrB[wave] = b; iscr[wave] = p; }
    }
    __syncthreads();
    if (tid == 0) {
        float a = 0.f, b = 0.f; int p = 0;
        for (int w = 0; w < NWAVES; ++w) { a += fscrA[w]; b += fscrB[w]; p += iscr[w]; }
        s_sl1 = a; s_poscon = b; s_pos = p;
        s_prefix = 0u;
        int nn = 3 * p; if (nn > NBOX) nn = NBOX;
        s_remaining = nn;
    }
    __syncthreads();

    const int pos = s_pos;
    if (pos == 0) {                 // num_mask == 0 -> row contributes 0
        if (tid == 0) row_out[n] = 0.f;
        return;                     // uniform exit for whole block
    }
    int neg_num = 3 * pos; if (neg_num > NBOX) neg_num = NBOX;

    // ---- 4-level radix select: t = neg_num-th largest key ----
    for (int level = 0; level < 4; ++level) {
        const int shiftDigit = 24 - 8 * level;
        if (level > 0) {
            __syncthreads();
            if (tid < 256) hist[tid] = 0u;
            const unsigned pmask  = 0xFFFFFFFFu << (32 - 8 * level);
            const unsigned prefix = s_prefix;
            __syncthreads();
            for (int q = tid; q < NBQ; q += TPB) {
                const uint4 kk = keys4[q];         // ds_load_b128
                const unsigned* kv = (const unsigned*)&kk;
#pragma unroll
                for (int j = 0; j < 4; ++j) {
                    const unsigned k = kv[j];
                    if ((k & pmask) == prefix)
                        atomicAdd(&hist[(k >> shiftDigit) & 255u], 1u);
                }
            }
            __syncthreads();
        }
        if (tid == 0) {
            int rem = s_remaining;
            unsigned cum = 0;
            int b = 0;
            for (int j = 255; j >= 0; --j) {
                const unsigned h = hist[j];
                if (cum + h >= (unsigned)rem) { b = j; s_remaining = rem - (int)cum; break; }
                cum += h;
            }
            s_prefix |= (unsigned)b << shiftDigit;
        }
        __syncthreads();
    }
    const unsigned t = s_prefix;

    // ---- count/sum of keys strictly above threshold ----
    float sum_gt = 0.f; int cnt_gt = 0;
    for (int q = tid; q < NBQ; q += TPB) {
        const uint4 kk = keys4[q];
        const unsigned* kv = (const unsigned*)&kk;
#pragma unroll
        for (int j = 0; j < 4; ++j) {
            const unsigned k = kv[j];
            if (k > t) { cnt_gt++; sum_gt += __uint_as_float(k); }
        }
    }
    {
        float a = waveSumF(sum_gt);
        int   c = waveSumI(cnt_gt);
        __syncthreads();
        if (lane == 0) { fscrA[wave] = a; iscr[wave] = c; }
    }
    __syncthreads();
    if (tid == 0) {
        float sgt = 0.f; int cgt = 0;
        for (int w = 0; w < NWAVES; ++w) { sgt += fscrA[w]; cgt += iscr[w]; }
        float remf = (float)(neg_num - cgt);
        if (remf < 0.f) remf = 0.f;  // tied keys have identical value -> sum invariant
        const float closs = s_poscon + sgt + remf * __uint_as_float(t);
        const float total = s_sl1 + closs;
        float posf = (float)pos;
        if (posf < 1e-6f) posf = 1e-6f;
        row_out[n] = total / posf;
    }
}

// ---- final mean over 256 rows via V_WMMA_F32_16X16X4_F32 ----
// A = ones(16x4) => D[m][n] = sum_k B[k][n] + C[m][n]; chained over 4 tiles
// of 64 values. Sum over D's 16 columns == sum of all 256 inputs, independent
// of the exact (k,n) placement of values into B's lane layout.
__global__ __launch_bounds__(32) void ssd_mean_wmma(
    const float* __restrict__ rows, float* __restrict__ out)
{
    const int lane = threadIdx.x;
    v8f c = {};
    v2f a; a.x = 1.0f; a.y = 1.0f;
    for (int tix = 0; tix < 4; ++tix) {
        v2f b;
        b.x = rows[tix * 64 + lane];
        b.y = rows[tix * 64 + 32 + lane];
        c = __builtin_amdgcn_wmma_f32_16x16x4_f32(
            /*neg_a=*/false, a, /*neg_b=*/false, b,
            /*c_mod=*/(short)0, c, /*reuse_a=*/false, /*reuse_b=*/false);
    }
    // lanes 0..15 of c[0] hold the 16 column sums (replicated in 16..31)
    float s = c[0];
    s += __shfl_xor(s, 1, 32);
    s += __shfl_xor(s, 2, 32);
    s += __shfl_xor(s, 4, 32);
    s += __shfl_xor(s, 8, 32);
    if (lane == 0) out[0] = s * (1.0f / (float)NBATCH);
}

extern "C" void kernel_launch(void* const* d_in, const int* in_sizes, int n_in,
                              void* d_out, int out_size, void* d_ws, size_t ws_size,
                              hipStream_t stream) {
    (void)in_sizes; (void)n_in; (void)out_size; (void)ws_size;
    const float* ploc   = (const float*)d_in[0];
    const float* plabel = (const float*)d_in[1];
    const float* gloc   = (const float*)d_in[2];
    const int*   glabel = (const int*)d_in[3];
    const float* dboxes = (const float*)d_in[4];
    float* rows = (float*)d_ws;   // NBATCH floats of scratch

    ssd_row_loss<<<NBATCH, TPB, 0, stream>>>(ploc, plabel, gloc, glabel, dboxes, rows);
    ssd_mean_wmma<<<1, 32, 0, stream>>>(rows, (float*)d_out);
}